// OTWBLoss_10952166604971
// MI455X (gfx1250) — compile-verified
//
#include <hip/hip_runtime.h>

// ---------------------------------------------------------------------------
// EMD between two 1D histograms: sum_i |cumsum(x0 - x1)[i]|, N = 2^23.
// Deterministic 2-sweep segmented scan; intra-tile (256-elem) scan done with
// V_WMMA_F32_16X16X4_F32 using triangular ones-matrices:
//    D  = L * B        (L lower-triangular incl.)  -> per-column prefixes
//    D += A2 * U       (U strictly upper, A2 = broadcast column totals)
// ---------------------------------------------------------------------------

typedef __attribute__((ext_vector_type(2))) float v2f;
typedef __attribute__((ext_vector_type(8))) float v8f;

#define N_TOTAL      8388608                    // 2^23
#define SEG          2048                       // elements per wave-segment
#define TILE         256                        // elements per WMMA tile
#define TILES_PER_SEG (SEG / TILE)              // 8
#define WAVES_PER_BLOCK 8
#define THREADS      256                        // 8 wave32
#define NSEG         (N_TOTAL / SEG)            // 4096
#define NBLK         (NSEG / WAVES_PER_BLOCK)   // 512

// ---------------- Pass 1: per-segment sums (one wave per 2048 elems) --------
__global__ __launch_bounds__(THREADS)
void k_segsum(const float* __restrict__ x0, const float* __restrict__ x1,
              float* __restrict__ segsum) {
  const int lane = threadIdx.x & 31;
  const int gw   = blockIdx.x * WAVES_PER_BLOCK + (threadIdx.x >> 5);
  const long base = (long)gw * SEG;
  float s = 0.f;
#pragma unroll 8
  for (int it = 0; it < SEG / 32; ++it) {
    long idx = base + (long)it * 32 + lane;     // coalesced 128B/instr
    s += x0[idx] - x1[idx];
  }
  for (int m = 16; m >= 1; m >>= 1) s += __shfl_xor(s, m, 32);
  if (lane == 0) segsum[gw] = s;
}

// ---------------- Pass 2: exclusive scan of 4096 segment sums (1 block) -----
__global__ __launch_bounds__(THREADS)
void k_scan(const float* __restrict__ segsum, float* __restrict__ segoff) {
  __shared__ float sh[THREADS];
  const int t = threadIdx.x;
  float v[NSEG / THREADS];                      // 16 values per thread
  float tot = 0.f;
#pragma unroll
  for (int i = 0; i < NSEG / THREADS; ++i) {
    v[i] = segsum[t * (NSEG / THREADS) + i];
    tot += v[i];
  }
  sh[t] = tot;
  __syncthreads();
  float inc = tot;                              // Hillis-Steele over threads
  for (int off = 1; off < THREADS; off <<= 1) {
    float add = (t >= off) ? sh[t - off] : 0.f;
    __syncthreads();
    inc += add;
    sh[t] = inc;
    __syncthreads();
  }
  float run = inc - tot;                        // exclusive across threads
#pragma unroll
  for (int i = 0; i < NSEG / THREADS; ++i) {
    segoff[t * (NSEG / THREADS) + i] = run;
    run += v[i];
  }
}

// ---------------- Pass 3: WMMA tile scans + |prefix| partials ---------------
__global__ __launch_bounds__(THREADS)
void k_emd(const float* __restrict__ x0, const float* __restrict__ x1,
           const float* __restrict__ segoff, float* __restrict__ blockpart) {
  const int lane = threadIdx.x & 31;
  const int wid  = threadIdx.x >> 5;
  const int gw   = blockIdx.x * WAVES_PER_BLOCK + wid;
  const int j    = lane & 15;                   // A-row (M) / B-column (N)
  const int dlt  = (lane >> 4) << 1;            // K sub-offset: 0 | 2

  // Constant triangular operands, laid out per ISA 32-bit A(16x4)/B(4x16):
  //   lanes 0-15: {K=4c, 4c+1}; lanes 16-31: {K=4c+2, 4c+3}
  v2f L[4], U[4];
#pragma unroll
  for (int c = 0; c < 4; ++c) {
    const int k0 = 4 * c + dlt;
    L[c].x = (k0     <= j) ? 1.f : 0.f;         // L[i][k] = (k <= i)
    L[c].y = (k0 + 1 <= j) ? 1.f : 0.f;
    U[c].x = (k0     <  j) ? 1.f : 0.f;         // U[k][j] = (k < j)
    U[c].y = (k0 + 1 <  j) ? 1.f : 0.f;
  }

  float carry = segoff[gw];                     // global exclusive prefix
  float labs  = 0.f;
  const long segbase = (long)gw * SEG;

  for (int tIdx = 0; tIdx < TILES_PER_SEG; ++tIdx) {
    const long tb = segbase + (long)tIdx * TILE;

    // B[k][j] = d[j*16 + k]; each lane loads 4 float2 pairs of its column.
    v2f B[4];
#pragma unroll
    for (int c = 0; c < 4; ++c) {
      const long off = tb + j * 16 + 4 * c + dlt;
      v2f a = *(const v2f*)(x0 + off);
      v2f b = *(const v2f*)(x1 + off);
      B[c] = a - b;
    }

    // D = L * B  (4 chained K=4 WMMAs) -> intra-column inclusive prefixes
    v8f acc = {0.f, 0.f, 0.f, 0.f, 0.f, 0.f, 0.f, 0.f};
#pragma unroll
    for (int c = 0; c < 4; ++c)
      acc = __builtin_amdgcn_wmma_f32_16x16x4_f32(
          false, L[c], false, B[c], (short)0, acc, false, false);

    // Column totals = C row M=15 -> VGPR 7, lanes 16..31 (N = lane-16).
    const float a7 = acc[7];

    // D += A2 * U : A2[i][k] = total_k (rows identical -> shfl broadcast)
#pragma unroll
    for (int c = 0; c < 4; ++c) {
      const int src = 16 + 4 * c + dlt;
      v2f A2;
      A2.x = __shfl(a7, src, 32);
      A2.y = __shfl(a7, src + 1, 32);
      acc = __builtin_amdgcn_wmma_f32_16x16x4_f32(
          false, A2, false, U[c], (short)0, acc, false, false);
    }

    // Full tile sum = element (M=15,N=15) = VGPR7 @ lane 31.
    const float tile_sum = __shfl(acc[7], 31, 32);
#pragma unroll
    for (int r = 0; r < 8; ++r) labs += fabsf(acc[r] + carry);
    carry += tile_sum;
  }

  for (int m = 16; m >= 1; m >>= 1) labs += __shfl_xor(labs, m, 32);

  __shared__ float red[WAVES_PER_BLOCK];
  if (lane == 0) red[wid] = labs;
  __syncthreads();
  if (threadIdx.x == 0) {
    float s = 0.f;
#pragma unroll
    for (int w = 0; w < WAVES_PER_BLOCK; ++w) s += red[w];
    blockpart[blockIdx.x] = s;
  }
}

// ---------------- Pass 4: reduce 512 block partials -> scalar ---------------
__global__ __launch_bounds__(THREADS)
void k_final(const float* __restrict__ blockpart, float* __restrict__ out) {
  __shared__ float sh[THREADS];
  const int t = threadIdx.x;
  sh[t] = blockpart[t] + blockpart[t + THREADS];   // NBLK = 512 = 2*THREADS
  __syncthreads();
  for (int off = THREADS / 2; off > 0; off >>= 1) {
    if (t < off) sh[t] += sh[t + off];
    __syncthreads();
  }
  if (t == 0) out[0] = sh[0];
}

// ---------------------------------------------------------------------------
extern "C" void kernel_launch(void* const* d_in, const int* in_sizes, int n_in,
                              void* d_out, int out_size, void* d_ws, size_t ws_size,
                              hipStream_t stream) {
  const float* x  = (const float*)d_in[0];    // [2, N] flat
  const float* x0 = x;
  const float* x1 = x + N_TOTAL;
  float* ws        = (float*)d_ws;
  float* segsum    = ws;                      // [NSEG]
  float* segoff    = ws + NSEG;               // [NSEG]
  float* blockpart = ws + 2 * NSEG;           // [NBLK]
  float* out       = (float*)d_out;

  k_segsum<<<NBLK, THREADS, 0, stream>>>(x0, x1, segsum);
  k_scan  <<<1,    THREADS, 0, stream>>>(segsum, segoff);
  k_emd   <<<NBLK, THREADS, 0, stream>>>(x0, x1, segoff, blockpart);
  k_final <<<1,    THREADS, 0, stream>>>(blockpart, out);
}